// HybridQuantumNet_88381837017833
// MI455X (gfx1250) — compile-verified
//
#include <hip/hip_runtime.h>
#include <hip/hip_bf16.h>

typedef __attribute__((ext_vector_type(2))) float v2f;
typedef __attribute__((ext_vector_type(8))) float v8f;

// ---------------------------------------------------------------------------
// Prologue: build the 16x16 complex unitary for the StronglyEntangling layers
// (weights-only, batch independent), lift it to the real 32x32 matrix
// A = [[Wr, -Wi], [Wi, Wr]] and store it directly in V_WMMA_F32_16X16X4_F32
// A-operand fragment layout:
//   frag[((t*8 + k)*32 + lane)*2 + j] = A[t*16 + lane%16][4k + j + 2*(lane/16)]
// ---------------------------------------------------------------------------
__global__ void build_unitary_kernel(const float* __restrict__ w,
                                     float* __restrict__ fragA) {
    __shared__ float Wre[16][16];
    __shared__ float Wim[16][16];
    const int tid = threadIdx.x;

    if (tid < 16) {
        // Simulate column `tid`: state = circuit_body @ e_tid  (16 complex regs)
        float sr[16], si[16];
#pragma unroll
        for (int m = 0; m < 16; ++m) { sr[m] = (m == tid) ? 1.f : 0.f; si[m] = 0.f; }

#pragma unroll
        for (int l = 0; l < 2; ++l) {
            // Rot(phi,theta,omega) on each wire q (wire 0 = MSB of state index)
#pragma unroll
            for (int q = 0; q < 4; ++q) {
                const float* wp = w + (l * 4 + q) * 3;
                float phi = wp[0], th = wp[1], om = wp[2];
                float ct, st, cap, sap, cam, sam;
                __sincosf(0.5f * th, &st, &ct);
                __sincosf(0.5f * (phi + om), &sap, &cap);
                __sincosf(0.5f * (phi - om), &sam, &cam);
                // m00 = e^{-i ap} ct ; m01 = -e^{+i am} st
                // m10 = e^{-i am} st ; m11 = e^{+i ap} ct
                float m00r =  ct * cap, m00i = -ct * sap;
                float m01r = -st * cam, m01i = -st * sam;
                float m10r =  st * cam, m10i = -st * sam;
                float m11r =  ct * cap, m11i =  ct * sap;
                const int stride = 1 << (3 - q);
#pragma unroll
                for (int m = 0; m < 16; ++m) {
                    if ((m & stride) == 0) {
                        int m1 = m | stride;
                        float a0r = sr[m],  a0i = si[m];
                        float a1r = sr[m1], a1i = si[m1];
                        sr[m]  = m00r*a0r - m00i*a0i + m01r*a1r - m01i*a1i;
                        si[m]  = m00r*a0i + m00i*a0r + m01r*a1i + m01i*a1r;
                        sr[m1] = m10r*a0r - m10i*a0i + m11r*a1r - m11i*a1i;
                        si[m1] = m10r*a0i + m10i*a0r + m11r*a1i + m11i*a1r;
                    }
                }
            }
            // CNOT ring: control q, target (q + r) % 4, r = l % 3 + 1
            const int r = (l % 3) + 1;
#pragma unroll
            for (int q = 0; q < 4; ++q) {
                int t  = (q + r) & 3;
                int cb = 1 << (3 - q);
                int tb = 1 << (3 - t);
#pragma unroll
                for (int m = 0; m < 16; ++m) {
                    if ((m & cb) && !(m & tb)) {
                        int m1 = m | tb;
                        float tr = sr[m]; sr[m] = sr[m1]; sr[m1] = tr;
                        float ti = si[m]; si[m] = si[m1]; si[m1] = ti;
                    }
                }
            }
        }
#pragma unroll
        for (int i = 0; i < 16; ++i) { Wre[i][tid] = sr[i]; Wim[i][tid] = si[i]; }
    }
    __syncthreads();

    // Emit A-operand fragments (1024 floats = 4 KB)
    for (int idx = tid; idx < 1024; idx += blockDim.x) {
        int j = idx & 1;
        int L = (idx >> 1) & 31;
        int k = (idx >> 6) & 7;
        int t = idx >> 9;
        int row = t * 16 + (L & 15);
        int col = 4 * k + j + 2 * (L >> 4);
        int ri = row & 15, ci = col & 15;
        float v;
        if (row < 16) v = (col < 16) ?  Wre[ri][ci] : -Wim[ri][ci];
        else          v = (col < 16) ?  Wim[ri][ci] :  Wre[ri][ci];
        fragA[idx] = v;
    }
}

// ---------------------------------------------------------------------------
// Main kernel: 8 waves / block, 16 batch elements per wave.
//   F[32,16] = A[32,32] x S[32,16] via 2 M-tiles x 8 K-steps of
//   V_WMMA_F32_16X16X4_F32 (fp32 in / fp32 accumulate).
// ---------------------------------------------------------------------------
__global__ __launch_bounds__(256) void qcirc_wmma_kernel(
    const float* __restrict__ x, const float* __restrict__ fragA,
    const float* __restrict__ fc_w, const float* __restrict__ fc_b,
    float* __restrict__ out, int nchunk) {

    __shared__ float sS[8][32][16];   // per-wave S: rows 0-15 Re, 16-31 Im

    const int lane  = threadIdx.x & 31;
    const int wave  = threadIdx.x >> 5;
    const int chunk = blockIdx.x * 8 + wave;
    const int e     = lane & 15;      // batch element within chunk
    const int half  = lane >> 4;      // 0: real rows, 1: imag rows

    const bool valid = (chunk < nchunk);   // wave-uniform

    if (valid) {
        const float4 xv = ((const float4*)x)[(size_t)chunk * 16 + e];
        float c0, s0, c1, s1, c2, s2, c3, s3;
        __sincosf(0.5f * xv.x, &s0, &c0);
        __sincosf(0.5f * xv.y, &s1, &c1);
        __sincosf(0.5f * xv.z, &s2, &c2);
        __sincosf(0.5f * xv.w, &s3, &c3);
        // Product-state amplitude m: magnitude * (-i)^popcount(m)
#pragma unroll
        for (int m = 0; m < 16; ++m) {
            float p = ((m & 8) ? s0 : c0) * ((m & 4) ? s1 : c1) *
                      ((m & 2) ? s2 : c2) * ((m & 1) ? s3 : c3);
            int pop = __popc(m);
            float v;
            if (half == 0)   // Re: nonzero iff pop even; sign -(pop%4==2)
                v = (pop & 1) ? 0.f : ((pop & 2) ? -p : p);
            else             // Im: nonzero iff pop odd; pop%4==1 -> -p, ==3 -> +p
                v = (pop & 1) ? ((pop & 2) ? p : -p) : 0.f;
            sS[wave][half * 16 + m][e] = v;
        }
    }
    __syncthreads();
    if (!valid) return;

    v8f accR = {};   // rows 0-15 of A*S  -> Re(final amplitudes)
    v8f accI = {};   // rows 16-31       -> Im(final amplitudes)
    const v2f* fA = (const v2f*)fragA;

#pragma unroll
    for (int k = 0; k < 8; ++k) {
        // B-fragment (4x16, K x N): VGPR j, lane L -> (K = 4k + j + 2*(L/16), N = L%16)
        v2f b;
        b.x = sS[wave][4 * k + 2 * half + 0][e];
        b.y = sS[wave][4 * k + 2 * half + 1][e];
        v2f a0 = fA[(0 * 8 + k) * 32 + lane];
        v2f a1 = fA[(1 * 8 + k) * 32 + lane];
        accR = __builtin_amdgcn_wmma_f32_16x16x4_f32(
            false, a0, false, b, (short)0, accR, false, false);
        accI = __builtin_amdgcn_wmma_f32_16x16x4_f32(
            false, a1, false, b, (short)0, accI, false, false);
    }

    // D layout: VGPR j, lane L -> amplitude m = j + 8*(L/16) of element L%16
    float z0 = 0.f, z1 = 0.f;
#pragma unroll
    for (int j = 0; j < 8; ++j) {
        int m = j + 8 * half;
        float pr = accR[j], pi = accI[j];
        float prob = pr * pr + pi * pi;
        z0 += (m & 8) ? -prob : prob;   // wire 0 = MSB
        z1 += (m & 4) ? -prob : prob;   // wire 1
    }
    // combine the two half-wave partial sums (different m ranges, same element)
    z0 += __shfl_xor(z0, 16);
    z1 += __shfl_xor(z1, 16);

    if (half == 0) {
        float o0 = z0 * fc_w[0] + z1 * fc_w[1] + fc_b[0];
        float o1 = z0 * fc_w[2] + z1 * fc_w[3] + fc_b[1];
        float2 o; o.x = o0; o.y = o1;
        ((float2*)out)[(size_t)chunk * 16 + e] = o;
    }
}

extern "C" void kernel_launch(void* const* d_in, const int* in_sizes, int n_in,
                              void* d_out, int out_size, void* d_ws, size_t ws_size,
                              hipStream_t stream) {
    const float* x    = (const float*)d_in[0];   // [B,4]
    const float* wts  = (const float*)d_in[1];   // [2,4,3]
    const float* fc_w = (const float*)d_in[2];   // [2,2]
    const float* fc_b = (const float*)d_in[3];   // [2]
    float*       out  = (float*)d_out;           // [B,2]
    float*       frag = (float*)d_ws;            // 4 KB A-fragments

    const int B      = in_sizes[0] / 4;
    const int nchunk = (B + 15) / 16;
    const int blocks = (nchunk + 7) / 8;

    build_unitary_kernel<<<1, 64, 0, stream>>>(wts, frag);
    qcirc_wmma_kernel<<<blocks, 256, 0, stream>>>(x, frag, fc_w, fc_b, out, nchunk);
}